// ModalityDropout_66185446031582
// MI455X (gfx1250) — compile-verified
//
#include <hip/hip_runtime.h>
#include <stdint.h>

// ---------------------------------------------------------------------------
// ModalityDropout for MI455X (gfx1250): pure streaming elementwise op.
// Bandwidth-bound (AI ~ 0.125 flop/byte) -> b128 NT loads/stores, wave32,
// block-uniform Bernoulli mask, scalar draws precomputed on host.
// ---------------------------------------------------------------------------

typedef __attribute__((ext_vector_type(4))) float float4v;

// Deterministic counter-based hash (stand-in for jax threefry; same
// Bernoulli semantics, deterministic across calls).
__host__ __device__ __forceinline__ uint32_t mix32(uint32_t x) {
    x ^= x >> 16; x *= 0x7feb352dU;
    x ^= x >> 15; x *= 0x846ca68bU;
    x ^= x >> 16;
    return x;
}
__host__ __device__ __forceinline__ float u01(uint32_t h) {
    // top 24 bits -> [0,1)
    return (float)(h >> 8) * (1.0f / 16777216.0f);
}

// One block per (b,t) row: 256 threads x float4 == 1024 floats == D.
// keep[b,t] is block-uniform -> computed once from blockIdx.x.
__global__ __launch_bounds__(256)
void md_audio_kernel(const float* __restrict__ audio,
                     const float* __restrict__ mask,
                     float* __restrict__ audio_out,
                     float* __restrict__ mask_out,
                     int vecPerRow,          // D/4
                     int globalKeep)         // 0/1 scalar from host
{
    const uint32_t row = blockIdx.x;         // row = b*T + t

    // per-(b,t) segment Bernoulli(0.10), keyed on seed 42
    const uint32_t hs = mix32(row * 0x9E3779B9u + 0x42424242u);
    const bool segDrop = u01(hs) < 0.10f;
    const float keep = (globalKeep && !segDrop) ? 1.0f : 0.0f;

    const size_t base = (size_t)row * (size_t)vecPerRow;
    const float4v* __restrict__ src = (const float4v*)audio + base;
    float4v* __restrict__ dst = (float4v*)audio_out + base;

    // Streaming pass: global_load_b128 / global_store_b128 with NT hint
    // (data touched exactly once; keep it out of L2's hot set).
    for (int i = threadIdx.x; i < vecPerRow; i += blockDim.x) {
        float4v v = __builtin_nontemporal_load(src + i);
        v *= keep;
        __builtin_nontemporal_store(v, dst + i);
    }

    if (threadIdx.x == 0) {
        mask_out[row] = mask[row] * keep;
    }
}

// vitals_out = vitals * vit_keep (scalar). 64 KiB of fp32 -> trivial.
__global__ __launch_bounds__(256)
void md_vitals_kernel(const float* __restrict__ vitals,
                      float* __restrict__ vitals_out,
                      int nVec,
                      float vitKeep)
{
    const int i = blockIdx.x * blockDim.x + threadIdx.x;
    if (i < nVec) {
        float4v v = __builtin_nontemporal_load((const float4v*)vitals + i);
        v *= vitKeep;
        __builtin_nontemporal_store(v, (float4v*)vitals_out + i);
    }
}

extern "C" void kernel_launch(void* const* d_in, const int* in_sizes, int n_in,
                              void* d_out, int out_size, void* d_ws, size_t ws_size,
                              hipStream_t stream)
{
    // Inputs (setup_inputs order): audio_tokens [B,T,D], vitals_token [B,D],
    // audio_mask [B,T] -- all fp32.
    const float* audio  = (const float*)d_in[0];
    const float* vitals = (const float*)d_in[1];
    const float* mask   = (const float*)d_in[2];

    const int nAudio = in_sizes[0];     // B*T*D
    const int nVit   = in_sizes[1];     // B*D
    const int nMask  = in_sizes[2];     // B*T
    const int rows   = nMask;           // B*T
    const int D      = nAudio / rows;   // 1024

    // Outputs concatenated flat in return order: audio_out, vitals_out, mask_out.
    float* audio_out  = (float*)d_out;
    float* vitals_out = audio_out + (size_t)nAudio;
    float* mask_out   = vitals_out + (size_t)nVit;

    // Scalar draws (deterministic, pure host arithmetic -> graph-capture safe):
    //   global_drop ~ Bernoulli(0.15), vit_drop ~ Bernoulli(0.20), seed 42.
    const uint32_t seed = 42u;
    const int   globalKeep = (u01(mix32(seed ^ 0x85ebca6bU)) < 0.15f) ? 0 : 1;
    const float vitKeep    = (u01(mix32(seed ^ 0x2545F491U)) >= 0.20f) ? 1.0f : 0.0f;

    // Audio + mask: one block per (b,t) row; 256 threads (8 wave32) x float4.
    const int vecPerRow = D / 4;
    md_audio_kernel<<<rows, 256, 0, stream>>>(audio, mask, audio_out, mask_out,
                                              vecPerRow, globalKeep);

    // Vitals: B*D/4 float4 lanes.
    const int nVec = nVit / 4;
    md_vitals_kernel<<<(nVec + 255) / 256, 256, 0, stream>>>(vitals, vitals_out,
                                                             nVec, vitKeep);
}